// ML3Encoder_57483842290077
// MI455X (gfx1250) — compile-verified
//
#include <hip/hip_runtime.h>
#include <hip/hip_bf16.h>

#define N_NODES  25000
#define N_EDGES  400000
#define N_GRAPHS 100
#define D_IN  16
#define DD    32
#define KS    11
#define NOUT1 30
#define NOUT2 2
#define LAYERS 2

typedef __attribute__((ext_vector_type(2))) float v2f;
typedef __attribute__((ext_vector_type(8))) float v8f;

// ---------- monotone float<->uint keys for atomic max over signed floats ----
__device__ __forceinline__ unsigned fkey(float x) {
  unsigned u = __float_as_uint(x);
  return (u & 0x80000000u) ? ~u : (u | 0x80000000u);
}
__device__ __forceinline__ float funkey(unsigned k) {
  return (k & 0x80000000u) ? __uint_as_float(k & 0x7fffffffu)
                           : __uint_as_float(~k);
}
#define MKEY_NEG_INF 0x007FFFFFu   /* fkey(-inf) */

// ---------------------------------------------------------------------------
// Generic f32 WMMA GEMM: C[M x ncols] = act(A[M x Kdim] @ B[Kdim x ncols] + bias) (+resid)
// One wave computes a 16-row x 32-col tile using V_WMMA_F32_16X16X4_F32.
// B is staged once per block into LDS, zero-padded to a fixed 32-col stride,
// so the inner loop is branch-free: global_load_b64 (A) + ds loads (B) + wmma.
// ncols <= 32; Kdim must be a multiple of 4; ldc = row stride of C/resid.
// Launch with dynamic shared mem = Kdim*32*4 bytes.
// ---------------------------------------------------------------------------
__global__ void wmma_gemm32(const float* __restrict__ A, int lda,
                            const float* __restrict__ B, int ncols,
                            const float* __restrict__ bias,
                            const float* __restrict__ resid,
                            float* __restrict__ Cout, int ldc,
                            int M, int Kdim, int relu_flag)
{
  extern __shared__ float sB[];            // [Kdim x 32], cols >= ncols are 0
  const int tid = threadIdx.x;
  for (int i = tid; i < (Kdim << 5); i += blockDim.x) {
    int kr = i >> 5, c = i & 31;
    sB[i] = (c < ncols) ? B[(size_t)kr * ncols + c] : 0.f;
  }
  __syncthreads();

  const int lane = tid & 31;
  const int wave = (blockIdx.x * blockDim.x + tid) >> 5;
  const int row0 = wave * 16;
  if (row0 >= M) return;                   // wave-uniform; after the barrier

  const int mrow = row0 + (lane & 15);
  const int mld  = (mrow < M) ? mrow : (M - 1);   // clamp; stores are guarded
  const int koff = (lane >> 4) * 2;               // lanes 0-15: K+0,1 ; 16-31: K+2,3
  const int ncol = lane & 15;

  v8f acc0 = {0.f,0.f,0.f,0.f,0.f,0.f,0.f,0.f};
  v8f acc1 = {0.f,0.f,0.f,0.f,0.f,0.f,0.f,0.f};

  const float* arow = A + (size_t)mld * lda + koff;
  for (int k = 0; k < Kdim; k += 4) {
    __builtin_prefetch(arow + k + 32, 0, 0);      // global_prefetch_b8
    v2f a; a.x = arow[k]; a.y = arow[k + 1];

    const float* bp = sB + ((k + koff) << 5);     // row k+koff of padded B
    v2f b0; b0.x = bp[ncol];      b0.y = bp[32 + ncol];
    v2f b1; b1.x = bp[ncol + 16]; b1.y = bp[32 + ncol + 16];

    acc0 = __builtin_amdgcn_wmma_f32_16x16x4_f32(false, a, false, b0,
                                                 (short)0, acc0, false, false);
    acc1 = __builtin_amdgcn_wmma_f32_16x16x4_f32(false, a, false, b1,
                                                 (short)0, acc1, false, false);
  }

  const int mbase = row0 + ((lane < 16) ? 0 : 8);
  for (int r = 0; r < 8; ++r) {
    int m = mbase + r;
    if (m >= M) continue;
    if (ncol < ncols) {
      float v = acc0[r];
      if (bias)      v += bias[ncol];
      if (relu_flag) v = v > 0.f ? v : 0.f;
      if (resid)     v += resid[(size_t)m * ldc + ncol];
      Cout[(size_t)m * ldc + ncol] = v;
    }
    int c1 = ncol + 16;
    if (c1 < ncols) {
      float v = acc1[r];
      if (bias)      v += bias[c1];
      if (relu_flag) v = v > 0.f ? v : 0.f;
      if (resid)     v += resid[(size_t)m * ldc + c1];
      Cout[(size_t)m * ldc + c1] = v;
    }
  }
}

// ---------------------------------------------------------------------------
// Per-layer scratch init: comb = 0, denom = 0, mkey = key(-inf)
// ---------------------------------------------------------------------------
__global__ void layer_init_kernel(float* __restrict__ comb,
                                  float* __restrict__ denom,
                                  unsigned* __restrict__ mkey)
{
  int gid = blockIdx.x * blockDim.x + threadIdx.x;
  const int total = N_NODES * KS * DD;        // 8,800,000
  if (gid < total) comb[gid] = 0.f;
  if (gid < N_NODES) { denom[gid] = 0.f; mkey[gid] = MKEY_NEG_INF; }
}

// ---------------------------------------------------------------------------
// Edge MLP: ea = relu(cat[relu(x@e1), tanh(x@e2)*tanh(x@e3)] @ e4)
// Weights cached in LDS (4.8 KB).
// ---------------------------------------------------------------------------
__global__ void edge_mlp_kernel(const float* __restrict__ edge_attr,
                                const float* __restrict__ e1w,
                                const float* __restrict__ e2w,
                                const float* __restrict__ e3w,
                                const float* __restrict__ e4w,
                                float* __restrict__ ea_out,
                                float* __restrict__ ea_mean)
{
  __shared__ float s1[KS * 2 * KS];   // 11 x 22
  __shared__ float s2[KS * 2 * KS];
  __shared__ float s3[KS * 2 * KS];
  __shared__ float s4[4 * KS * KS];   // 44 x 11
  for (int i = threadIdx.x; i < KS * 2 * KS; i += blockDim.x) {
    s1[i] = e1w[i]; s2[i] = e2w[i]; s3[i] = e3w[i];
  }
  for (int i = threadIdx.x; i < 4 * KS * KS; i += blockDim.x) s4[i] = e4w[i];
  __syncthreads();

  int e = blockIdx.x * blockDim.x + threadIdx.x;
  if (e >= N_EDGES) return;

  float a[KS];
#pragma unroll
  for (int i = 0; i < KS; ++i) a[i] = edge_attr[(size_t)e * KS + i];

  float cat[4 * KS];                  // [lin(22) | gate(22)]
#pragma unroll
  for (int j = 0; j < 2 * KS; ++j) {
    float s = 0.f, t = 0.f, u = 0.f;
#pragma unroll
    for (int i = 0; i < KS; ++i) {
      float ai = a[i];
      s += ai * s1[i * 2 * KS + j];
      t += ai * s2[i * 2 * KS + j];
      u += ai * s3[i * 2 * KS + j];
    }
    cat[j]          = s > 0.f ? s : 0.f;
    cat[2 * KS + j] = tanhf(t) * tanhf(u);
  }
  float sum = 0.f;
#pragma unroll
  for (int j = 0; j < KS; ++j) {
    float o = 0.f;
#pragma unroll
    for (int i = 0; i < 4 * KS; ++i) o += cat[i] * s4[i * KS + j];
    o = o > 0.f ? o : 0.f;
    ea_out[(size_t)e * KS + j] = o;
    sum += o;
  }
  ea_mean[e] = sum * (1.0f / KS);
}

// ---------------------------------------------------------------------------
// raw attention + segment max (monotone-uint atomicMax)
// raw = ea_mean[e] * dot(h_proj[src], att_v)
// ---------------------------------------------------------------------------
__global__ void raw_att_kernel(const float* __restrict__ hproj,
                               const float* __restrict__ ea_mean,
                               const float* __restrict__ att_v,
                               const int* __restrict__ src,
                               const int* __restrict__ dst,
                               float* __restrict__ rawbuf,
                               unsigned* __restrict__ mkey)
{
  int e = blockIdx.x * blockDim.x + threadIdx.x;
  if (e >= N_EDGES) return;
  int s = src[e];
  float dotv = 0.f;
#pragma unroll
  for (int d = 0; d < DD; ++d) dotv += hproj[(size_t)s * DD + d] * att_v[d];
  float raw = ea_mean[e] * dotv;
  rawbuf[e] = raw;
  atomicMax(&mkey[dst[e]], fkey(raw));
}

// ---------------------------------------------------------------------------
// e = exp(raw - m[dst]) (m := 0 when -inf), denom[dst] += e ; rawbuf <- e
// ---------------------------------------------------------------------------
__global__ void seg_exp_kernel(const int* __restrict__ dst,
                               const unsigned* __restrict__ mkey,
                               float* __restrict__ rawbuf,
                               float* __restrict__ denom)
{
  int e = blockIdx.x * blockDim.x + threadIdx.x;
  if (e >= N_EDGES) return;
  int d = dst[e];
  float m = funkey(mkey[d]);
  if (m < -3.0e38f) m = 0.f;          // only -inf possible here
  float ex = expf(rawbuf[e] - m);
  rawbuf[e] = ex;
  atomicAdd(&denom[d], ex);
}

// ---------------------------------------------------------------------------
// Aggregation (wave-per-edge, lane = feature d):
//   att = e/(denom+eps); atts_out[e] = att
//   comb[dst,k,d] += (1+att) * ea[e,k] * h_proj[src,d]
// (sum_out + att_out == einsum(agg + att_agg, conv_w) by linearity)
// ---------------------------------------------------------------------------
__global__ void aggregate_kernel(const float* __restrict__ hproj,
                                 const float* __restrict__ ea,
                                 const float* __restrict__ ebuf,
                                 const float* __restrict__ denom,
                                 const int* __restrict__ src,
                                 const int* __restrict__ dst,
                                 float* __restrict__ comb,
                                 float* __restrict__ atts_out)
{
  const int lane   = threadIdx.x & 31;
  const int wave   = (blockIdx.x * blockDim.x + threadIdx.x) >> 5;
  const int nwaves = (gridDim.x * blockDim.x) >> 5;
  for (int e = wave; e < N_EDGES; e += nwaves) {
    int s = src[e], d = dst[e];
    float att = ebuf[e] / (denom[d] + 1e-16f);
    if (lane == 0) atts_out[e] = att;
    float xs = hproj[(size_t)s * DD + lane];
    float w  = (1.0f + att) * xs;
    float* cbase = comb + (size_t)d * (KS * DD) + lane;
#pragma unroll
    for (int k = 0; k < KS; ++k)
      atomicAdd(cbase + k * DD, w * ea[(size_t)e * KS + k]);
  }
}

// ---------------------------------------------------------------------------
// skip = tanh(h_proj@s1+b1) * tanh(h_proj@s2+b2) -> layer_out cols 30..31
// ---------------------------------------------------------------------------
__global__ void skip_kernel(const float* __restrict__ hproj,
                            const float* __restrict__ s1w, const float* __restrict__ s1b,
                            const float* __restrict__ s2w, const float* __restrict__ s2b,
                            float* __restrict__ layer_out)
{
  int n = blockIdx.x * blockDim.x + threadIdx.x;
  if (n >= N_NODES) return;
#pragma unroll
  for (int j = 0; j < NOUT2; ++j) {
    float a = s1b[j], b = s2b[j];
#pragma unroll
    for (int d = 0; d < DD; ++d) {
      float hv = hproj[(size_t)n * DD + d];
      a += hv * s1w[d * NOUT2 + j];
      b += hv * s2w[d * NOUT2 + j];
    }
    layer_out[(size_t)n * DD + NOUT1 + j] = tanhf(a) * tanhf(b);
  }
}

// ---------------------------------------------------------------------------
// Global mean pool
// ---------------------------------------------------------------------------
__global__ void pool_init_kernel(float* __restrict__ psum, float* __restrict__ cnt)
{
  int gid = blockIdx.x * blockDim.x + threadIdx.x;
  if (gid < N_GRAPHS * DD) psum[gid] = 0.f;
  if (gid < N_GRAPHS)      cnt[gid]  = 0.f;
}

__global__ void pool_acc_kernel(const float* __restrict__ h,
                                const int* __restrict__ batch,
                                float* __restrict__ psum, float* __restrict__ cnt)
{
  const int lane = threadIdx.x & 31;
  const int wave = (blockIdx.x * blockDim.x + threadIdx.x) >> 5;
  if (wave >= N_NODES) return;
  int b = batch[wave];
  atomicAdd(&psum[(size_t)b * DD + lane], h[(size_t)wave * DD + lane]);
  if (lane == 0) atomicAdd(&cnt[b], 1.0f);
}

__global__ void pool_div_kernel(const float* __restrict__ psum,
                                const float* __restrict__ cnt,
                                float* __restrict__ pooled)
{
  int gid = blockIdx.x * blockDim.x + threadIdx.x;
  if (gid >= N_GRAPHS * DD) return;
  float c = cnt[gid / DD];
  pooled[gid] = psum[gid] / (c > 1.0f ? c : 1.0f);
}

// ---------------------------------------------------------------------------
extern "C" void kernel_launch(void* const* d_in, const int* in_sizes, int n_in,
                              void* d_out, int out_size, void* d_ws, size_t ws_size,
                              hipStream_t stream)
{
  // ---- inputs (setup_inputs order) ----
  const float* x        = (const float*)d_in[0];
  const float* edge_attr= (const float*)d_in[1];
  const int*   eidx     = (const int*)  d_in[2];
  const int*   batch    = (const int*)  d_in[3];
  const float* emb_w    = (const float*)d_in[4];
  const float* emb_b    = (const float*)d_in[5];
  const float* init_w   = (const float*)d_in[6];
  const float* e1_w     = (const float*)d_in[7];
  const float* e2_w     = (const float*)d_in[8];
  const float* e3_w     = (const float*)d_in[9];
  const float* e4_w     = (const float*)d_in[10];
  const float* conv_w   = (const float*)d_in[11];
  const float* conv_b   = (const float*)d_in[12];
  const float* att_v    = (const float*)d_in[13];
  const float* s1_w     = (const float*)d_in[14];
  const float* s1_b     = (const float*)d_in[15];
  const float* s2_w     = (const float*)d_in[16];
  const float* s2_b     = (const float*)d_in[17];
  const float* out_w    = (const float*)d_in[18];
  const float* out_b    = (const float*)d_in[19];
  const float* final_w  = (const float*)d_in[20];
  const float* final_b  = (const float*)d_in[21];

  const int* src = eidx;
  const int* dst = eidx + N_EDGES;

  // ---- outputs: [output 100x32 | h 25000x32 | atts 2x400000] ----
  float* out_g  = (float*)d_out;
  float* h      = out_g + N_GRAPHS * DD;
  float* atts   = h + (size_t)N_NODES * DD;

  // ---- workspace layout (floats) ----
  float* ws = (float*)d_ws;
  float*    hproj   = ws;                                  // 800,000
  float*    layerout= hproj   + (size_t)N_NODES * DD;      // 800,000
  float*    ea      = layerout+ (size_t)N_NODES * DD;      // 4,400,000
  float*    ea_mean = ea      + (size_t)N_EDGES * KS;      // 400,000
  float*    rawbuf  = ea_mean + N_EDGES;                   // 400,000
  float*    denom   = rawbuf  + N_EDGES;                   // 25,000
  unsigned* mkey    = (unsigned*)(denom + N_NODES);        // 25,000
  float*    comb    = (float*)(mkey + N_NODES);            // 8,800,000
  float*    psum    = comb + (size_t)N_NODES * KS * DD;    // 3,200
  float*    cnt     = psum + N_GRAPHS * DD;                // 100
  float*    pooled  = cnt + N_GRAPHS;                      // 3,200
  (void)ws_size; (void)in_sizes; (void)n_in; (void)out_size;

  const int TB = 256;
  const int edge_blocks = (N_EDGES + TB - 1) / TB;            // 1563
  const int node_blocks = (N_NODES + TB - 1) / TB;            // 98
  const int init_blocks = (N_NODES * KS * DD + TB - 1) / TB;  // 34375
  auto gemm_grid = [](int M) { int nt = (M + 15) / 16; return (nt + 7) / 8; };
  auto shb       = [](int Kdim) { return (size_t)Kdim * 32 * sizeof(float); };

  // h = x @ emb_w + emb_b
  wmma_gemm32<<<gemm_grid(N_NODES), TB, shb(D_IN), stream>>>(
      x, D_IN, emb_w, DD, emb_b, nullptr, h, DD, N_NODES, D_IN, 0);

  for (int l = 0; l < LAYERS; ++l) {
    const float* iw  = init_w + (size_t)l * DD * DD;
    const float* w1  = e1_w   + (size_t)l * KS * 2 * KS;
    const float* w2  = e2_w   + (size_t)l * KS * 2 * KS;
    const float* w3  = e3_w   + (size_t)l * KS * 2 * KS;
    const float* w4  = e4_w   + (size_t)l * 4 * KS * KS;
    const float* cw  = conv_w + (size_t)l * KS * DD * NOUT1;  // [352 x 30]
    const float* cb  = conv_b + (size_t)l * NOUT1;
    const float* av  = att_v  + (size_t)l * DD;
    const float* a1w = s1_w   + (size_t)l * DD * NOUT2;
    const float* a1b = s1_b   + (size_t)l * NOUT2;
    const float* a2w = s2_w   + (size_t)l * DD * NOUT2;
    const float* a2b = s2_b   + (size_t)l * NOUT2;
    const float* ow  = out_w  + (size_t)l * DD * DD;
    const float* ob  = out_b  + (size_t)l * DD;
    float* atts_l = atts + (size_t)l * N_EDGES;

    // h_proj = h @ init_w[l]
    wmma_gemm32<<<gemm_grid(N_NODES), TB, shb(DD), stream>>>(
        h, DD, iw, DD, nullptr, nullptr, hproj, DD, N_NODES, DD, 0);

    // reset comb / denom / mkey
    layer_init_kernel<<<init_blocks, TB, 0, stream>>>(comb, denom, mkey);

    // edge MLP
    edge_mlp_kernel<<<edge_blocks, TB, 0, stream>>>(
        edge_attr, w1, w2, w3, w4, ea, ea_mean);

    // raw attention + segment max
    raw_att_kernel<<<edge_blocks, TB, 0, stream>>>(
        hproj, ea_mean, av, src, dst, rawbuf, mkey);

    // exp + denom
    seg_exp_kernel<<<edge_blocks, TB, 0, stream>>>(dst, mkey, rawbuf, denom);

    // combined aggregation (+ write atts)
    aggregate_kernel<<<8192, TB, 0, stream>>>(
        hproj, ea, rawbuf, denom, src, dst, comb, atts_l);

    // conv_out = relu(comb @ conv_w + conv_b) -> layer_out[:, 0:30]
    wmma_gemm32<<<gemm_grid(N_NODES), TB, shb(KS * DD), stream>>>(
        comb, KS * DD, cw, NOUT1, cb, nullptr, layerout, DD, N_NODES, KS * DD, 1);

    // skip -> layer_out[:, 30:32]
    skip_kernel<<<node_blocks, TB, 0, stream>>>(
        hproj, a1w, a1b, a2w, a2b, layerout);

    // h = h + layer_out @ out_w + out_b
    wmma_gemm32<<<gemm_grid(N_NODES), TB, shb(DD), stream>>>(
        layerout, DD, ow, DD, ob, h, h, DD, N_NODES, DD, 0);
  }

  // global mean pool + final head
  pool_init_kernel<<<(N_GRAPHS * DD + TB - 1) / TB, TB, 0, stream>>>(psum, cnt);
  pool_acc_kernel<<<(N_NODES * 32 + TB - 1) / TB, TB, 0, stream>>>(h, batch, psum, cnt);
  pool_div_kernel<<<(N_GRAPHS * DD + TB - 1) / TB, TB, 0, stream>>>(psum, cnt, pooled);

  wmma_gemm32<<<gemm_grid(N_GRAPHS), TB, shb(DD), stream>>>(
      pooled, DD, final_w, DD, final_b, nullptr, out_g, DD, N_GRAPHS, DD, 0);
}